// MultiplexController_58763742544155
// MI455X (gfx1250) — compile-verified
//
#include <hip/hip_runtime.h>
#include <stdint.h>

// Problem shape fixed by the reference: B=4096 objects, C=4096 channels,
// M*S = 4096 (bucket, slot) entries. out[b,:] = count[b] * x[b,:].
#define B_DIM  4096
#define C_DIM  4096
#define MS_DIM 4096

typedef uint32_t u32x4 __attribute__((ext_vector_type(4)));
typedef uint32_t u32x8 __attribute__((ext_vector_type(8)));

__global__ __launch_bounds__(256) void mux_zero_counts(int* __restrict__ counts, int n) {
    int i = blockIdx.x * blockDim.x + threadIdx.x;
    if (i < n) counts[i] = 0;
}

__global__ __launch_bounds__(256) void mux_histogram(const int* __restrict__ assignments,
                                                     int* __restrict__ counts,
                                                     int n, int b_dim) {
    int i = blockIdx.x * blockDim.x + threadIdx.x;
    if (i < n) {
        int a = assignments[i];
        if (a >= 0 && a < b_dim) atomicAdd(&counts[a], 1);
    }
}

// One block per row b. Stage the 16KB row into LDS:
//   - first 8KB via the Tensor Data Mover (one tensor_load_to_lds, TENSORcnt)
//   - second 8KB via per-lane async global->LDS copies (ASYNCcnt)
// then scale by count[b] and stream out.
__global__ __launch_bounds__(256) void mux_scale_rows(const float* __restrict__ x,
                                                      const int*   __restrict__ counts,
                                                      const int*   __restrict__ assignments,
                                                      float*       __restrict__ out) {
    __shared__ float tile[C_DIM];          // 16 KB of the 320 KB/WGP LDS
    const int b = blockIdx.x;
    const int t = threadIdx.x;

    const float* row = x + (size_t)b * C_DIM;

    // Low 32 bits of the generic shared-space address are the workgroup-
    // relative LDS byte offset (aperture bits live in [63:32]).
    uint32_t lds_base = (uint32_t)(uintptr_t)tile;

    // ---- First half (cols 0..2047): single TDM DMA issued by wave 0 ----
    if (t < 32) {
        uint64_t ga = (uint64_t)(uintptr_t)row;
        u32x4 g0;
        g0.x = 1u;                                   // count=1, user descriptor
        g0.y = lds_base;                             // lds_addr (bytes)
        g0.z = (uint32_t)ga;                         // global_addr[31:0]
        g0.w = ((uint32_t)(ga >> 32) & 0x01FFFFFFu)  // global_addr[56:32]
             | (2u << 30);                           // type = 2 ("image")
        u32x8 g1;
        g1.s0 = 2u << 16;      // workgroup_mask=0 (no cluster), data_size=2 (4B)
        g1.s1 = 2048u << 16;   // tensor_dim0 = 2048 elements (low 16 bits)
        g1.s2 = 1u << 16;      // tensor_dim0 hi = 0, tensor_dim1 = 1
        g1.s3 = 2048u << 16;   // tile_dim0 = 2048 elements (8 KB, 1-D tile)
        g1.s4 = 0u;            // tile_dim1 = 0 (unused), tile_dim2 = 0
        g1.s5 = 2048u;         // tensor_dim0_stride = 2048
        g1.s6 = 0u;
        g1.s7 = 0u;
        asm volatile("tensor_load_to_lds %0, %1" :: "s"(g0), "s"(g1) : "memory");
    }

    // ---- Second half (cols 2048..4095): per-lane async b128 copies ----
#pragma unroll
    for (int i = 2; i < 4; ++i) {
        int idx = i * 256 + t;                       // 512..1023 (float4 index)
        uint32_t lds_addr = lds_base + (uint32_t)(idx * 16);
        uint64_t gaddr    = (uint64_t)(uintptr_t)(row + idx * 4);
        asm volatile("global_load_async_to_lds_b128 %0, %1, off"
                     :: "v"(lds_addr), "v"(gaddr)
                     : "memory");
    }

    // Fetch the scalar multiplicity while the copies are in flight.
    float scale;
    if (counts) {
        scale = (float)counts[b];           // s_load, block-uniform
    } else {
        // Fallback (no workspace): count occurrences of b in assignments.
        __shared__ int cnt;
        if (t == 0) cnt = 0;
        __syncthreads();
        int local = 0;
        for (int i = t; i < MS_DIM; i += 256)
            local += (assignments[i] == b) ? 1 : 0;
        if (local) atomicAdd(&cnt, local);
        __syncthreads();
        scale = (float)cnt;
    }

    // Async half: each lane reads back exactly the bytes it targeted,
    // so the wave-local ASYNCcnt wait is sufficient for that half.
    asm volatile("s_wait_asynccnt 0" ::: "memory");
    // TDM half: wave 0 waits for the DMA, then barrier for cross-wave
    // LDS visibility of the first 8 KB.
    if (t < 32) __builtin_amdgcn_s_wait_tensorcnt(0);
    __syncthreads();

    const float4* tile4 = (const float4*)tile;
    float4*       out4  = (float4*)(out + (size_t)b * C_DIM);
#pragma unroll
    for (int i = 0; i < 4; ++i) {
        int idx = i * 256 + t;
        float4 v = tile4[idx];             // ds_load_b128
        v.x *= scale; v.y *= scale; v.z *= scale; v.w *= scale;
        out4[idx] = v;                     // global_store_b128
    }
}

extern "C" void kernel_launch(void* const* d_in, const int* in_sizes, int n_in,
                              void* d_out, int out_size, void* d_ws, size_t ws_size,
                              hipStream_t stream) {
    const float* x           = (const float*)d_in[0];
    const int*   assignments = (const int*)d_in[1];
    float*       out         = (float*)d_out;

    const int MS = in_sizes[1];   // 4096 (M*S)
    const int B  = B_DIM;

    int* counts = nullptr;
    if (ws_size >= (size_t)B * sizeof(int)) {
        counts = (int*)d_ws;
        mux_zero_counts<<<(B  + 255) / 256, 256, 0, stream>>>(counts, B);
        mux_histogram  <<<(MS + 255) / 256, 256, 0, stream>>>(assignments, counts, MS, B);
    }
    mux_scale_rows<<<B_DIM, 256, 0, stream>>>(x, counts, assignments, out);
    (void)n_in; (void)out_size;
}